// FB6NLLLoss_30923764531211
// MI455X (gfx1250) — compile-verified
//
#include <hip/hip_runtime.h>

#define NQ 256
#define NBLOCKS 2048
#define PI_D 3.14159265358979323846
#define LOG2E 1.4426950408889634f

typedef __attribute__((ext_vector_type(2))) float v2f;
typedef __attribute__((ext_vector_type(4))) float v4f;
typedef __attribute__((ext_vector_type(8))) float v8f;

// ---------------------------------------------------------------------------
// Kernel 1: Gauss-Legendre nodes/weights (Numerical Recipes gauleg), f64.
// ---------------------------------------------------------------------------
__global__ void gauleg_kernel(float* __restrict__ gU, float* __restrict__ gS2,
                              float* __restrict__ gLW, int n) {
  int i = blockIdx.x * blockDim.x + threadIdx.x;
  if (i >= n) return;
  double x = cos(PI_D * (i + 0.75) / (n + 0.5));
  double pp = 1.0;
  for (int it = 0; it < 64; ++it) {
    double p1 = 1.0, p2 = 0.0;
    for (int j = 1; j <= n; ++j) {
      double p3 = p2; p2 = p1;
      p1 = ((2.0 * j - 1.0) * x * p2 - (j - 1.0) * p3) / (double)j;
    }
    pp = n * (x * p1 - p2) / (x * x - 1.0);
    double dx = p1 / pp;
    x -= dx;
    if (fabs(dx) < 1e-14) break;
  }
  double w = 2.0 / ((1.0 - x * x) * pp * pp);
  gU[i]  = (float)x;
  gS2[i] = (float)(1.0 - x * x);
  gLW[i] = (float)log(w);
}

// exp2(T2) * I0(z), where T2 is the log2e-scaled linear term (from WMMA) and
// z in [0,~8]. Branchless; raw v_rcp_f32 / v_rsq_f32 / v_exp_f32 (no OCML
// denorm guards -- z >= 3.75 whenever rcp/rsq results are consumed).
__device__ __forceinline__ float expT_I0(float T2, float z) {
  // small: I0(z) = Ps(t^2), t = z/3.75
  float t2 = z * (1.0f / 3.75f); t2 *= t2;
  float ps = 1.0f + t2 * (3.5156229f + t2 * (3.0899424f + t2 * (1.2067492f +
             t2 * (0.2659732f + t2 * (0.0360768f + t2 * 0.0045813f)))));
  // large: I0(z) = e^z * Pl(3.75/z) / sqrt(z)
  float ti = 3.75f * __builtin_amdgcn_rcpf(z);
  float pl = 0.39894228f + ti * (0.01328592f + ti * (0.00225319f + ti * (-0.00157565f +
             ti * (0.00916281f + ti * (-0.02057706f + ti * (0.02635537f +
             ti * (-0.01647633f + ti * 0.00392377f)))))));
  pl *= __builtin_amdgcn_rsqf(z);
  bool lg = z >= 3.75f;
  float zl  = lg ? LOG2E : 0.0f;          // large branch adds z*log2e to exponent
  float mul = lg ? pl : ps;
  return __builtin_amdgcn_exp2f(fmaf(z, zl, T2)) * mul;
}

// ---------------------------------------------------------------------------
// Kernel 2: main FB6 NLL. Each wave owns 16 points; 16 x V_WMMA_F32_16X16X4_F32
// cover 256 quadrature nodes. WMMA computes the log2e-scaled linear term
//   T2[m,n] = log2e * (u_m*kappa_n + s2_m*a_n + logw_m - (kappa_n+beta_n))
// (shift M = kappa+beta is a valid logsumexp bound since a+b = beta and
//  log I0(z) <= z). VALU adds exp2(T2)*I0(b*s2) per element.
// ---------------------------------------------------------------------------
__global__ void fb6_kernel(const float* __restrict__ xv,
                           const float* __restrict__ th_, const float* __restrict__ ph_,
                           const float* __restrict__ ps_, const float* __restrict__ kap_,
                           const float* __restrict__ bet_, const float* __restrict__ eta_,
                           const float* __restrict__ gU, const float* __restrict__ gS2,
                           const float* __restrict__ gLW,
                           float* __restrict__ partials, int Bn) {
  __shared__ float2 tabAB[2 * NQ];   // [0..NQ): {u, s2}; [NQ..2NQ): {logw, 1}
  __shared__ float  sS2[NQ];
  __shared__ float  sRed[256];
  for (int i = threadIdx.x; i < NQ; i += blockDim.x) {
    float u = gU[i], s2 = gS2[i], lw = gLW[i];
    sS2[i] = s2;
    tabAB[i]      = make_float2(u,  s2);
    tabAB[i + NQ] = make_float2(lw, 1.0f);
  }
  __syncthreads();

  const int lane = threadIdx.x & 31;
  const int wid  = threadIdx.x >> 5;
  const int wavesPerBlock = blockDim.x >> 5;
  const int gwave  = blockIdx.x * wavesPerBlock + wid;
  const int nwaves = gridDim.x * wavesPerBlock;
  const int nGroups = Bn >> 4;           // 16 points per wave-group
  const bool hi = lane >= 16;
  const int  n  = lane & 15;             // point index within group
  const int  hioff  = hi ? NQ : 0;       // A-table select, no branch
  const int  mbase0 = hi ? 8  : 0;       // D-row base for this half-wave

  float acc = 0.0f;
  for (int g = gwave; g < nGroups; g += nwaves) {   // uniform across wave
    const int pt = (g << 4) + n;
    const float kap = kap_[pt];
    const float bet = bet_[pt];
    const float et  = eta_[pt];
    const float aa = 0.5f * bet * (1.0f - et);
    const float bb = 0.5f * bet * (1.0f + et);

    // B fragment (4 x 16 points), K-rows scaled by log2e:
    //   kappa*L, a*L, L, -(kappa+beta)*L
    v2f Bm;
    Bm.x = (hi ? 1.0f : kap) * LOG2E;
    Bm.y = (hi ? -(kap + bet) : aa) * LOG2E;

    float S = 0.0f;
    #pragma unroll 2
    for (int t = 0; t < 16; ++t) {
      const int m = (t << 4) + n;            // A-row (node) for this lane
      const float2 ab = tabAB[m + hioff];    // one ds_load_b64
      v2f Am; Am.x = ab.x; Am.y = ab.y;
      v8f T = {0.f,0.f,0.f,0.f,0.f,0.f,0.f,0.f};
      T = __builtin_amdgcn_wmma_f32_16x16x4_f32(
              false, Am, false, Bm, (short)0, T, false, false);
      const int mb = (t << 4) + mbase0;      // 8-aligned -> b128 loads
      const v4f* s2p = reinterpret_cast<const v4f*>(&sS2[mb]);
      const v4f s2a = s2p[0];
      const v4f s2b = s2p[1];
      S += expT_I0(T[0], bb * s2a[0]);
      S += expT_I0(T[1], bb * s2a[1]);
      S += expT_I0(T[2], bb * s2a[2]);
      S += expT_I0(T[3], bb * s2a[3]);
      S += expT_I0(T[4], bb * s2b[0]);
      S += expT_I0(T[5], bb * s2b[1]);
      S += expT_I0(T[6], bb * s2b[2]);
      S += expT_I0(T[7], bb * s2b[3]);
    }
    // combine node-halves: lanes n and n+16 hold complementary node sets
    S += __shfl_xor(S, 16, 32);

    // per-point term1 = kappa*g1 + beta*(g2^2 - eta*g3^2), Gamma = H(th,ph)K(ps)
    float st, ct, sp, cp, ss, cs;
    __sincosf(th_[pt], &st, &ct);
    __sincosf(ph_[pt], &sp, &cp);
    __sincosf(ps_[pt], &ss, &cs);
    const float x0 = xv[pt * 3 + 0], x1 = xv[pt * 3 + 1], x2 = xv[pt * 3 + 2];
    const float g1 = x0 * ct + x1 * st * cp + x2 * st * sp;
    const float g2 = -x0 * st * cs + x1 * (ct * cp * cs - sp * ss)
                                   + x2 * (ct * sp * cs + cp * ss);
    const float g3 =  x0 * st * ss + x1 * (-ct * cp * ss - sp * cs)
                                   + x2 * (-ct * sp * ss + cp * cs);
    const float term1 = kap * g1 + bet * (g2 * g2 - et * g3 * g3);
    const float logC = 1.8378770664093453f /* log(2*pi) */ + kap + bet + __logf(S);
    acc += hi ? 0.0f : (logC - term1);   // cndmask, keeps EXEC all-ones
  }

  // block reduction -> one partial per block (deterministic, no atomics)
  sRed[threadIdx.x] = acc;
  __syncthreads();
  for (int s = blockDim.x >> 1; s > 0; s >>= 1) {
    if (threadIdx.x < s) sRed[threadIdx.x] += sRed[threadIdx.x + s];
    __syncthreads();
  }
  if (threadIdx.x == 0) partials[blockIdx.x] = sRed[0];
}

// ---------------------------------------------------------------------------
// Kernel 3: final reduction of per-block partials -> d_out[0] = sum / B
// ---------------------------------------------------------------------------
__global__ void reduce_kernel(const float* __restrict__ partials, int nPart,
                              float* __restrict__ out, float invB) {
  __shared__ float sRed[256];
  float a = 0.0f;
  for (int i = threadIdx.x; i < nPart; i += blockDim.x) a += partials[i];
  sRed[threadIdx.x] = a;
  __syncthreads();
  for (int s = blockDim.x >> 1; s > 0; s >>= 1) {
    if (threadIdx.x < s) sRed[threadIdx.x] += sRed[threadIdx.x + s];
    __syncthreads();
  }
  if (threadIdx.x == 0) out[0] = sRed[0] * invB;
}

extern "C" void kernel_launch(void* const* d_in, const int* in_sizes, int n_in,
                              void* d_out, int out_size, void* d_ws, size_t ws_size,
                              hipStream_t stream) {
  (void)n_in; (void)out_size; (void)ws_size;
  const float* x     = (const float*)d_in[0];
  const float* theta = (const float*)d_in[1];
  const float* phi   = (const float*)d_in[2];
  const float* psi   = (const float*)d_in[3];
  const float* kappa = (const float*)d_in[4];
  const float* beta  = (const float*)d_in[5];
  const float* eta   = (const float*)d_in[6];
  const int Bn = in_sizes[1];

  float* wsf      = (float*)d_ws;
  float* gU       = wsf;
  float* gS2      = wsf + NQ;
  float* gLW      = wsf + 2 * NQ;
  float* partials = wsf + 3 * NQ;

  gauleg_kernel<<<1, NQ, 0, stream>>>(gU, gS2, gLW, NQ);
  fb6_kernel<<<NBLOCKS, 256, 0, stream>>>(x, theta, phi, psi, kappa, beta, eta,
                                          gU, gS2, gLW, partials, Bn);
  reduce_kernel<<<1, 256, 0, stream>>>(partials, NBLOCKS, (float*)d_out,
                                       1.0f / (float)Bn);
}